// RandomWalkPixelLabeling_48146583388201
// MI455X (gfx1250) — compile-verified
//
#include <hip/hip_runtime.h>
#include <stdint.h>

#define IMG  1024
#define NBLK 1024   // (1024/32)^2 blocks

// ---------------- CDNA5 async global->LDS helpers (gfx1250) ----------------
// Per-lane 16-byte global->LDS DMA, tracked by ASYNCcnt (CDNA5 ISA).
__device__ __forceinline__ void async_b128_to_lds(unsigned lds_byte_addr,
                                                  const void* gaddr) {
  asm volatile("global_load_async_to_lds_b128 %0, %1, off"
               :: "v"(lds_byte_addr), "v"(gaddr)
               : "memory");
}

__device__ __forceinline__ void s_wait_asynccnt_0() {
#if __has_builtin(__builtin_amdgcn_s_wait_asynccnt)
  __builtin_amdgcn_s_wait_asynccnt(0);
#else
  asm volatile("s_wait_asynccnt 0" ::: "memory");
#endif
}

__device__ __forceinline__ float wave_sum(float v) {
#pragma unroll
  for (int off = 16; off > 0; off >>= 1) v += __shfl_down(v, off, 32);
  return v;
}

// ---------------- init: stencil P, one-hot x0, uint8 seed class ----------------
__global__ __launch_bounds__(256) void rw_init(const float* __restrict__ img,
                                               const int* __restrict__ seeds,
                                               float4* __restrict__ P,
                                               float2* __restrict__ x0,
                                               unsigned char* __restrict__ cls) {
  int idx = blockIdx.x * 256 + threadIdx.x;
  int i = idx >> 10;
  int j = idx & 1023;
  float c = img[idx];
  bool cb = c > 0.1f;
  const int di[4] = {-1, 1, 0, 0};   // DIRS order: up, down, left, right
  const int dj[4] = { 0, 0,-1, 1};
  float w[4];
  float rs = 0.f;
#pragma unroll
  for (int d = 0; d < 4; ++d) {
    int ni = i + di[d], nj = j + dj[d];
    bool v = ((unsigned)ni < IMG) && ((unsigned)nj < IMG);
    float nb = v ? img[ni * IMG + nj] : 0.f;
    float same = (cb != (nb > 0.1f)) ? 1.f : 0.f;
    float inten = fabsf(c - nb + same);
    float wd = v ? 0.36787944117144233f * expf(-inten) : 0.f;  // exp(-1)*exp(-I)
    w[d] = wd;
    rs += wd;
  }
  if (rs == 0.f) rs = 1.f;
  P[idx] = make_float4(w[0] / rs, w[1] / rs, w[2] / rs, w[3] / rs);  // match ref div
  int s = seeds[idx];
  cls[idx] = (unsigned char)s;
  x0[idx] = make_float2(s == 1 ? 1.f : 0.f, s == 2 ? 1.f : 0.f);
}

// ---------------- two fused Jacobi steps per launch ----------------
// Owned tile 32x32; stage 36x36 (halo 2) of x; intermediate 34x34 in LDS.
__global__ __launch_bounds__(256) void rw_fused(
    const float2* __restrict__ xin, float2* __restrict__ xout,
    const float4* __restrict__ P, const unsigned char* __restrict__ cls,
    const float* __restrict__ ppA, const float* __restrict__ ppB,
    float* __restrict__ cpA, float* __restrict__ cpB, int first) {
  __shared__ __align__(16) float2 xs[36][36];   // staged x (halo 2)
  __shared__ __align__(16) float2 xm[34][34];   // intermediate (halo 1)
  __shared__ float wsA[8], wsB[8];

  const int tx = threadIdx.x, ty = threadIdx.y;       // block = (32, 8)
  const int tid = ty * 32 + tx;
  const int lane = tid & 31, wid = tid >> 5;
  const int gx0 = blockIdx.x * 32, gy0 = blockIdx.y * 32;
  const int blk = blockIdx.y * 32 + blockIdx.x;

  // (1) Kick off 36x36 tile staging: 648 async 16B (2-pixel) DMAs.
#pragma unroll
  for (int r = 0; r < 3; ++r) {
    int pi = tid + 256 * r;
    if (pi < 648) {
      int si = pi / 18;
      int sj = (pi - si * 18) * 2;       // even -> 16B aligned pairs
      int gi = gy0 + si - 2, gj = gx0 + sj - 2;
      if (((unsigned)gi < IMG) && ((unsigned)gj < IMG)) {
        async_b128_to_lds((unsigned)(uintptr_t)&xs[si][sj],
                          (const void*)(xin + gi * IMG + gj));
      } else {
        *(float4*)&xs[si][sj] = make_float4(0.f, 0.f, 0.f, 0.f);  // zero pad
      }
    }
  }

  // (2) Convergence state from previous fused kernel (overlaps the DMA).
  float accA = 0.f, accB = 0.f;
  if (!first) {
#pragma unroll
    for (int r = 0; r < 4; ++r) {
      accA += ppA[tid + 256 * r];
      accB += ppB[tid + 256 * r];
    }
  }
  accA = wave_sum(accA);
  accB = wave_sum(accB);
  if (lane == 0) { wsA[wid] = accA; wsB[wid] = accB; }
  __syncthreads();
  float rA = 0.f, rB = 0.f;
#pragma unroll
  for (int w = 0; w < 8; ++w) { rA += wsA[w]; rB += wsB[w]; }
  const bool frozen = (!first) && ((rA < 1e-8f) || (rB < 1e-8f));

  if (frozen) {   // sticky copy-through; partials stay 0 so freeze persists
#pragma unroll
    for (int k = 0; k < 4; ++k) {
      int gi = gy0 + 4 * ty + k, gj = gx0 + tx;
      xout[gi * IMG + gj] = xin[gi * IMG + gj];
    }
    if (tid == 0) { cpA[blk] = 0.f; cpB[blk] = 0.f; }
    return;
  }

  // Owned-cell P / seed class live in registers across both half-steps.
  float4 p[4];
  int cl[4];
#pragma unroll
  for (int k = 0; k < 4; ++k) {
    int gi = gy0 + 4 * ty + k, gj = gx0 + tx;
    p[k]  = P[gi * IMG + gj];
    cl[k] = cls[gi * IMG + gj];
  }

  // (3) Drain ASYNCcnt, publish tile.
  s_wait_asynccnt_0();
  __syncthreads();

  // (4) Half-step A: 34x34 intermediate into xm. Owned cells + residual A.
  float resA = 0.f;
  float2 a[4];
  {
    const int c = tx;
    float2 colS[6];
#pragma unroll
    for (int r = 0; r < 6; ++r) colS[r] = xs[4 * ty + 1 + r][c + 2];
#pragma unroll
    for (int k = 0; k < 4; ++k) {
      int t = 4 * ty + k;
      float2 up = colS[k], ct = colS[k + 1], dn = colS[k + 2];
      float2 lf = xs[t + 2][c + 1], rt = xs[t + 2][c + 3];
      float nx = p[k].x * up.x + p[k].y * dn.x + p[k].z * lf.x + p[k].w * rt.x;
      float ny = p[k].x * up.y + p[k].y * dn.y + p[k].z * lf.y + p[k].w * rt.y;
      if (cl[k] > 0) { nx = (cl[k] == 1) ? 1.f : 0.f; ny = (cl[k] == 2) ? 1.f : 0.f; }
      a[k] = make_float2(nx, ny);
      xm[t + 1][c + 1] = a[k];
      float dx = nx - ct.x, dy = ny - ct.y;
      resA += dx * dx + dy * dy;
    }
  }
  // Ring cells of xm (owned by neighbor blocks -> no residual contribution).
  if (tid < 132) {
    int mi, mj;
    if (tid < 34)       { mi = 0;            mj = tid; }
    else if (tid < 68)  { mi = 33;           mj = tid - 34; }
    else if (tid < 100) { mi = tid - 68 + 1; mj = 0; }
    else                { mi = tid - 100 + 1; mj = 33; }
    int gi = gy0 + mi - 1, gj = gx0 + mj - 1;
    float2 v = make_float2(0.f, 0.f);
    if (((unsigned)gi < IMG) && ((unsigned)gj < IMG)) {
      float4 pr = P[gi * IMG + gj];
      int cr = cls[gi * IMG + gj];
      float2 up = xs[mi][mj + 1], dn = xs[mi + 2][mj + 1];
      float2 lf = xs[mi + 1][mj], rt = xs[mi + 1][mj + 2];
      v.x = pr.x * up.x + pr.y * dn.x + pr.z * lf.x + pr.w * rt.x;
      v.y = pr.x * up.y + pr.y * dn.y + pr.z * lf.y + pr.w * rt.y;
      if (cr > 0) { v.x = (cr == 1) ? 1.f : 0.f; v.y = (cr == 2) ? 1.f : 0.f; }
    }
    xm[mi][mj] = v;
  }
  __syncthreads();

  // (5) Half-step B over owned 32x32: write out + residual B.
  float resB = 0.f;
  {
    const int c = tx;
    float2 colM[6];
#pragma unroll
    for (int r = 0; r < 6; ++r) colM[r] = xm[4 * ty + r][c + 1];
#pragma unroll
    for (int k = 0; k < 4; ++k) {
      int t = 4 * ty + k;
      float2 up = colM[k], dn = colM[k + 2];
      float2 lf = xm[t + 1][c], rt = xm[t + 1][c + 2];
      float nx = p[k].x * up.x + p[k].y * dn.x + p[k].z * lf.x + p[k].w * rt.x;
      float ny = p[k].x * up.y + p[k].y * dn.y + p[k].z * lf.y + p[k].w * rt.y;
      if (cl[k] > 0) { nx = (cl[k] == 1) ? 1.f : 0.f; ny = (cl[k] == 2) ? 1.f : 0.f; }
      int gi = gy0 + t, gj = gx0 + c;
      xout[gi * IMG + gj] = make_float2(nx, ny);
      float dx = nx - a[k].x, dy = ny - a[k].y;
      resB += dx * dx + dy * dy;
    }
  }

  // (6) Deterministic block partials (wave32 shuffle + tiny LDS stage).
  resA = wave_sum(resA);
  resB = wave_sum(resB);
  __syncthreads();                       // protect wsA/wsB reuse
  if (lane == 0) { wsA[wid] = resA; wsB[wid] = resB; }
  __syncthreads();
  if (tid == 0) {
    float sa = 0.f, sb = 0.f;
#pragma unroll
    for (int w = 0; w < 8; ++w) { sa += wsA[w]; sb += wsB[w]; }
    cpA[blk] = sa;
    cpB[blk] = sb;
  }
}

// ---------------- launch ----------------
extern "C" void kernel_launch(void* const* d_in, const int* in_sizes, int n_in,
                              void* d_out, int out_size, void* d_ws, size_t ws_size,
                              hipStream_t stream) {
  (void)in_sizes; (void)n_in; (void)out_size; (void)ws_size;
  const float* image = (const float*)d_in[0];
  const int*   seeds = (const int*)d_in[1];

  // Workspace layout (~33.1 MB):
  //   [0,16MB)    P    : float4 per pixel
  //   [16,24MB)   buf0 : float2 per pixel (x ping)
  //   [24,32MB)   buf1 : float2 per pixel (x pong)
  //   [32,33MB)   cls  : uint8 per pixel
  //   [33MB,..)   partials: 4 x 1024 floats (A0, A1, B0, B1)
  char* ws = (char*)d_ws;
  float4*        P    = (float4*)(ws);
  float2*        buf0 = (float2*)(ws + ((size_t)16 << 20));
  float2*        buf1 = (float2*)(ws + ((size_t)24 << 20));
  unsigned char* cl   = (unsigned char*)(ws + ((size_t)32 << 20));
  float*         pA0  = (float*)(ws + ((size_t)33 << 20));
  float*         pA1  = pA0 + NBLK;
  float*         pB0  = pA1 + NBLK;
  float*         pB1  = pB0 + NBLK;

  rw_init<<<dim3(4096), dim3(256), 0, stream>>>(image, seeds, P, buf0, cl);

  dim3 grid(32, 32), blk(32, 8);
  for (int k = 1; k <= 50; ++k) {              // kernel k = reference steps 2k-1, 2k
    const float2* xin = (k & 1) ? buf0 : buf1;
    float2*       out = (k == 50) ? (float2*)d_out : ((k & 1) ? buf1 : buf0);
    const float*  ppA = ((k - 1) & 1) ? pA1 : pA0;
    const float*  ppB = ((k - 1) & 1) ? pB1 : pB0;
    float*        cpA = (k & 1) ? pA1 : pA0;
    float*        cpB = (k & 1) ? pB1 : pB0;
    rw_fused<<<grid, blk, 0, stream>>>(xin, out, P, cl, ppA, ppB, cpA, cpB,
                                       (k == 1) ? 1 : 0);
  }
}